// MultiGPULoss_71176198029451
// MI455X (gfx1250) — compile-verified
//
#include <hip/hip_runtime.h>
#include <math.h>

typedef float v2f __attribute__((ext_vector_type(2)));
typedef float v8f __attribute__((ext_vector_type(8)));

#define BB 4
#define NP 4096
#define MP 2048
#define W_CHAMFER 1.0f
#define W_REP 0.1f
#define W_COV 0.2f
#define W_SMOOTH 0.05f
#define REP_THRESHOLD 0.005f

__device__ __forceinline__ float guard_sqrt(float sq) {
  sq = fmaxf(sq, 0.0f);
  return (sq > 1e-12f) ? sqrtf(sq) : 0.0f;
}

// ---------------------------------------------------------------------------
__global__ void zero_accums_k(float* acc) {
  acc[0] = 0.0f;
  acc[1] = 0.0f;
}

// out[i] = x^2 + y^2 + z^2 for point i
__global__ void norms_k(const float* __restrict__ pts, float* __restrict__ out, int total) {
  int i = blockIdx.x * blockDim.x + threadIdx.x;
  if (i < total) {
    float x = pts[3 * i + 0];
    float y = pts[3 * i + 1];
    float z = pts[3 * i + 2];
    out[i] = x * x + y * y + z * z;
  }
}

// ---------------------------------------------------------------------------
// For each row m of A (BB batches, Na points each), compute min over n of
// squared distance to B[n] (Nb points per batch) via V_WMMA_F32_16X16X4_F32
// on the A.B^T term (K=4, z-pad 0).  One wave owns a 16-row tile and loops
// over Nb in 2x16-column tiles.  Branch-free fragment construction keeps
// EXEC all-1s (required by WMMA) and the loop body straight-line.
// Stores raw min squared distance (sqrt + 1e-12 guard deferred: both are
// monotone, so min commutes with them).
__global__ void __launch_bounds__(256)
rowmin_wmma_k(const float* __restrict__ A, const float* __restrict__ Bp,
              const float* __restrict__ nA, const float* __restrict__ nB,
              float* __restrict__ rowmin, int Na, int Nb) {
  const int wave = (blockIdx.x * blockDim.x + threadIdx.x) >> 5;
  const int lane = threadIdx.x & 31;
  const int tilesPerBatch = Na >> 4;
  const int b = wave / tilesPerBatch;
  const int rowbase = (wave % tilesPerBatch) << 4;
  const bool hi = (lane >> 4) != 0;   // lanes 16-31 carry the K=2 (z) half
  const int l16 = lane & 15;

  // A fragment (16x4 fp32): lanes 0-15 -> (x,y) of row l16;
  // lanes 16-31 -> (z, 0) of row l16.  Branchless: load all 3, select.
  const float* ap = A + ((size_t)b * Na + rowbase + l16) * 3;
  const float ax = ap[0], ay = ap[1], az = ap[2];
  v2f afrag;
  afrag.x = hi ? az : ax;
  afrag.y = hi ? 0.0f : ay;

  // Row norms for the 8 rows this lane-half covers in the C layout
  // (C VGPR i: lanes 0-15 hold row i, lanes 16-31 hold row i+8).
  float narow[8];
#pragma unroll
  for (int i = 0; i < 8; ++i)
    narow[i] = nA[(size_t)b * Na + rowbase + (hi ? 8 : 0) + i];

  float rmin[8];
#pragma unroll
  for (int i = 0; i < 8; ++i) rmin[i] = 3.0e38f;

  const float* bbase = Bp + (size_t)b * Nb * 3;
  const float* nbbase = nB + (size_t)b * Nb;

  for (int nt = 0; nt < Nb; nt += 32) {
#pragma unroll
    for (int u = 0; u < 2; ++u) {
      const int n = nt + u * 16 + l16;
      const float* bp = bbase + (size_t)n * 3;
      const float bx = bp[0], by = bp[1], bz = bp[2];
      v2f bfrag;
      bfrag.x = hi ? bz : bx;
      bfrag.y = hi ? 0.0f : by;
      const float nbcol = nbbase[n];   // norm of this lane's column point

      v8f c = {0.f, 0.f, 0.f, 0.f, 0.f, 0.f, 0.f, 0.f};
      c = __builtin_amdgcn_wmma_f32_16x16x4_f32(
          /*neg_a=*/false, afrag, /*neg_b=*/false, bfrag,
          /*c_mod=*/(short)0, c, /*reuse_a=*/false, /*reuse_b=*/false);

#pragma unroll
      for (int i = 0; i < 8; ++i) {
        float sq = narow[i] + nbcol - 2.0f * c[i];
        rmin[i] = fminf(rmin[i], sq);
      }
    }
  }

  // Min across the 16 lanes of each half (columns) -> per-row min.
#pragma unroll
  for (int i = 0; i < 8; ++i) {
    float v = rmin[i];
    v = fminf(v, __shfl_xor(v, 1, 32));
    v = fminf(v, __shfl_xor(v, 2, 32));
    v = fminf(v, __shfl_xor(v, 4, 32));
    v = fminf(v, __shfl_xor(v, 8, 32));
    rmin[i] = v;
  }
  if (l16 == 0) {
#pragma unroll
    for (int i = 0; i < 8; ++i)
      rowmin[(size_t)b * Na + rowbase + (hi ? 8 : 0) + i] = rmin[i];
  }
}

// ---------------------------------------------------------------------------
// One thread per pred point: maintain the 16 smallest squared distances
// (including self, d=0) with indices via a fully unrolled register insertion
// sort.  Candidate points are staged through LDS in 256-point tiles (the
// whole block scans the same candidates; LDS reads are broadcast).
// Repulsion uses the 8 smallest excluding self; smoothness uses all 16
// neighbor coordinates.  Wave-reduced, then one atomicAdd per wave.
__global__ void __launch_bounds__(256)
knn_k(const float* __restrict__ pred, float* __restrict__ acc, int N) {
  __shared__ float shx[256], shy[256], shz[256];

  const int tid = blockIdx.x * blockDim.x + threadIdx.x;   // covers BB*N exactly
  const int b = tid / N;                                   // uniform per block
  const int m = tid - b * N;
  const float* base = pred + (size_t)b * N * 3;
  const float px = base[3 * m + 0];
  const float py = base[3 * m + 1];
  const float pz = base[3 * m + 2];

  float d[16];
  int idx[16];
#pragma unroll
  for (int j = 0; j < 16; ++j) { d[j] = 3.0e38f; idx[j] = -1; }

  for (int nt = 0; nt < N; nt += 256) {
    __syncthreads();
    {
      const float* q = base + (size_t)(nt + threadIdx.x) * 3;
      shx[threadIdx.x] = q[0];
      shy[threadIdx.x] = q[1];
      shz[threadIdx.x] = q[2];
    }
    __syncthreads();
    for (int j = 0; j < 256; ++j) {
      const float dx = shx[j] - px;
      const float dy = shy[j] - py;
      const float dz = shz[j] - pz;
      const float sq = dx * dx + dy * dy + dz * dz;
      if (sq < d[15]) {
        d[15] = sq; idx[15] = nt + j;
#pragma unroll
        for (int k = 15; k > 0; --k) {
          if (d[k] < d[k - 1]) {
            float td = d[k]; d[k] = d[k - 1]; d[k - 1] = td;
            int ti = idx[k]; idx[k] = idx[k - 1]; idx[k - 1] = ti;
          }
        }
      }
    }
  }

  // Repulsion: 8 nearest excluding self (list ascending; self has d=0).
  float rep = 0.0f;
  int taken = 0;
#pragma unroll
  for (int j = 0; j < 16; ++j) {
    if (idx[j] != m && taken < 8) {
      float dist = guard_sqrt(d[j]);
      rep += fmaxf(REP_THRESHOLD - dist, 0.0f);
      ++taken;
    }
  }

  // Smoothness: variance of the 16 nearest (incl. self) neighbor coords.
  float sx = 0.f, sy = 0.f, sz = 0.f, ss = 0.f;
#pragma unroll
  for (int j = 0; j < 16; ++j) {
    const float* q = base + (size_t)idx[j] * 3;
    float qx = q[0], qy = q[1], qz = q[2];
    sx += qx; sy += qy; sz += qz;
    ss += qx * qx + qy * qy + qz * qz;
  }
  // sum of squared deviations = sum(q^2) - |sum(q)|^2 / 16
  float var_raw = ss - (sx * sx + sy * sy + sz * sz) * (1.0f / 16.0f);

  // wave32 reduction, one atomic per wave
  float r = rep, v = var_raw;
#pragma unroll
  for (int msk = 16; msk >= 1; msk >>= 1) {
    r += __shfl_xor(r, msk, 32);
    v += __shfl_xor(v, msk, 32);
  }
  if ((threadIdx.x & 31) == 0) {
    atomicAdd(&acc[0], r);
    atomicAdd(&acc[1], v);
  }
}

// ---------------------------------------------------------------------------
__device__ float block_sum_256(float v) {
  __shared__ float smem[8];
#pragma unroll
  for (int msk = 16; msk >= 1; msk >>= 1) v += __shfl_xor(v, msk, 32);
  const int lane = threadIdx.x & 31;
  const int w = threadIdx.x >> 5;
  __syncthreads();
  if (lane == 0) smem[w] = v;
  __syncthreads();
  float r = 0.0f;
  if (threadIdx.x == 0) {
#pragma unroll
    for (int i = 0; i < 8; ++i) r += smem[i];
  }
  return r;   // valid in thread 0 only
}

__global__ void __launch_bounds__(256)
finalize_k(const float* __restrict__ rm_pg, const float* __restrict__ rm_gp,
           const float* __restrict__ rm_cp, const float* __restrict__ partial,
           const float* __restrict__ acc, float* __restrict__ out,
           int N, int M) {
  const int tid = threadIdx.x;

  float s1 = 0.f, s2 = 0.f;
  for (int i = tid; i < BB * N; i += 256) {
    s1 += guard_sqrt(rm_pg[i]);
    s2 += guard_sqrt(rm_gp[i]);
  }
  const float sum_pg = block_sum_256(s1);
  const float sum_gp = block_sum_256(s2);

  float cov_mean = 0.0f;
  for (int b = 0; b < BB; ++b) {
    float sc = 0.f, sn = 0.f;
    for (int i = tid; i < M; i += 256) {
      const float* pp = partial + ((size_t)b * M + i) * 3;
      const float am = fabsf(pp[0]) + fabsf(pp[1]) + fabsf(pp[2]);
      const float mask = (am > 1e-6f) ? 1.0f : 0.0f;
      sc += mask * guard_sqrt(rm_cp[(size_t)b * M + i]);
      sn += mask;
    }
    const float scs = block_sum_256(sc);
    const float sns = block_sum_256(sn);
    if (tid == 0) cov_mean += (sns > 0.0f) ? scs / fmaxf(sns, 1.0f) : 0.0f;
  }

  if (tid == 0) {
    const float chamfer   = (sum_pg + sum_gp) / (float)(BB * N) * W_CHAMFER;
    const float repulsion = acc[0] / (float)(BB * N * 8) * W_REP;
    const float coverage  = (cov_mean / (float)BB) * W_COV;
    const float smooth    = (acc[1] / 47.0f) / (float)(BB * N) * W_SMOOTH;
    out[0] = chamfer;
    out[1] = repulsion;
    out[2] = coverage;
    out[3] = smooth;
    out[4] = chamfer + repulsion + coverage + smooth;
  }
}

// ---------------------------------------------------------------------------
extern "C" void kernel_launch(void* const* d_in, const int* in_sizes, int n_in,
                              void* d_out, int out_size, void* d_ws, size_t ws_size,
                              hipStream_t stream) {
  (void)in_sizes; (void)n_in; (void)out_size; (void)ws_size;
  const float* pred    = (const float*)d_in[0];   // 4 x 4096 x 3
  const float* gt      = (const float*)d_in[1];   // 4 x 4096 x 3
  const float* partial = (const float*)d_in[2];   // 4 x 2048 x 3
  float* out = (float*)d_out;                      // 5 floats

  float* ws      = (float*)d_ws;
  float* rm_pg   = ws;                       // BB*NP
  float* rm_gp   = rm_pg + BB * NP;          // BB*NP
  float* rm_cp   = rm_gp + BB * NP;          // BB*MP
  float* n_pred  = rm_cp + BB * MP;          // BB*NP
  float* n_gt    = n_pred + BB * NP;         // BB*NP
  float* n_part  = n_gt + BB * NP;           // BB*MP
  float* accums  = n_part + BB * MP;         // 2

  zero_accums_k<<<1, 1, 0, stream>>>(accums);

  norms_k<<<(BB * NP + 255) / 256, 256, 0, stream>>>(pred, n_pred, BB * NP);
  norms_k<<<(BB * NP + 255) / 256, 256, 0, stream>>>(gt, n_gt, BB * NP);
  norms_k<<<(BB * MP + 255) / 256, 256, 0, stream>>>(partial, n_part, BB * MP);

  // one wave per 16-row tile; 8 waves per 256-thread block
  const int blk_pg = (BB * (NP / 16)) / 8;   // 128
  const int blk_cp = (BB * (MP / 16)) / 8;   // 64
  rowmin_wmma_k<<<blk_pg, 256, 0, stream>>>(pred, gt, n_pred, n_gt, rm_pg, NP, NP);
  rowmin_wmma_k<<<blk_pg, 256, 0, stream>>>(gt, pred, n_gt, n_pred, rm_gp, NP, NP);
  rowmin_wmma_k<<<blk_cp, 256, 0, stream>>>(partial, pred, n_part, n_pred, rm_cp, MP, NP);

  knn_k<<<(BB * NP) / 256, 256, 0, stream>>>(pred, accums, NP);

  finalize_k<<<1, 256, 0, stream>>>(rm_pg, rm_gp, rm_cp, partial, accums, out, NP, MP);
}